// RenONet_65000035058177
// MI455X (gfx1250) — compile-verified
//
#include <hip/hip_runtime.h>
#include <cstddef>

// ---------------- problem constants ----------------
#define NN      5000
#define NPK     5008          // K padding (B rows, G rows, S rows used as K)
#define NPM     5120          // M padding for 128-row blocked GEMMs over nodes
#define NEDGE   80000
#define FIN     32
#define DD      64
#define P1N     256
#define P2N     32
#define MTOT    5288          // N + P1 + P2
#define MPAD    5376          // MTOT padded to 128
#define HH      256
#define BCOLS   320           // 64 (E1) + 256 (S)
#define GCOLS   384           // 64 (ze) + 256 (AS) + 1 (y) + zero pad to 64-multiple
#define G2COLS  80            // 64 (ze2) + 1 (y1) + pad
#define KC      32            // K-chunk staged in LDS
#define LOG_SCALE 1.4763057f

typedef __attribute__((ext_vector_type(2))) float v2f;
typedef __attribute__((ext_vector_type(8))) float v8f;

#define HAVE_ASYNC_LDS 0
#if defined(__has_builtin)
#if __has_builtin(__builtin_amdgcn_global_load_async_to_lds_b32) && \
    __has_builtin(__builtin_amdgcn_s_wait_asynccnt)
#undef HAVE_ASYNC_LDS
#define HAVE_ASYNC_LDS 1
#endif
#endif

// ---------------- utility kernels ----------------
__global__ void k_zero(float* __restrict__ p, size_t n) {
  size_t i = (size_t)blockIdx.x * blockDim.x + threadIdx.x;
  size_t stride = (size_t)gridDim.x * blockDim.x;
  for (; i < n; i += stride) p[i] = 0.0f;
}

__global__ void k_scatter_edges(const int* __restrict__ adj, float* __restrict__ A) {
  int e = blockIdx.x * 256 + threadIdx.x;
  if (e >= NEDGE) return;
  int i = adj[e];
  int j = adj[NEDGE + e];
  A[(size_t)i * NPK + j] = 1.0f;   // idempotent: duplicates still give 1.0
}

// z[i,d] = tanh(sum_k x0[i,k]*Wenc[k,d]) * LOG_SCALE, d < 64 (cols 64..66 unused)
__global__ void k_encoder(const float* __restrict__ x0, const float* __restrict__ Wenc,
                          float* __restrict__ Z) {
  int i = blockIdx.x;
  int d = threadIdx.x;
  float s = 0.f;
#pragma unroll
  for (int k = 0; k < FIN; ++k) s += x0[i * FIN + k] * Wenc[k * (DD + 3) + d];
  Z[(size_t)i * DD + d] = tanhf(s) * LOG_SCALE;
}

// ---------------- generic one-wave-per-tile WMMA GEMM (small/odd shapes) ----------------
// V_WMMA_F32_16X16X4_F32 layouts (ISA 7.12.2):
//   A 16x4 : lane(m=lane&15), v0=A[m,k0+2*hi], v1=A[m,k0+2*hi+1]   (hi = lane>>4)
//   B 4x16 : lane(n=lane&15), v0=B[k0+2*hi,n], v1=B[k0+2*hi+1,n]
//   C 16x16: vgpr r -> row m0 + r + 8*hi, col n0 + (lane&15)
template <bool TRANS_A>
__global__ void k_wmma_gemm(const float* __restrict__ A, int lda,
                            const float* __restrict__ B, int ldb,
                            float* __restrict__ C, int ldc,
                            int K, float alpha, const float* __restrict__ bias) {
  const int m0 = blockIdx.x * 16;
  const int n0 = blockIdx.y * 16;
  const int lane = threadIdx.x;
  const int lm = lane & 15;
  const int hi = lane >> 4;
  v8f acc = {};
  for (int k = 0; k < K; k += 4) {
    const int ka = k + 2 * hi;
    v2f a, b;
    if (TRANS_A) {
      a.x = A[(size_t)ka * lda + (m0 + lm)];
      a.y = A[(size_t)(ka + 1) * lda + (m0 + lm)];
    } else {
      a.x = A[(size_t)(m0 + lm) * lda + ka];
      a.y = A[(size_t)(m0 + lm) * lda + ka + 1];
    }
    b.x = B[(size_t)ka * ldb + (n0 + lm)];
    b.y = B[(size_t)(ka + 1) * ldb + (n0 + lm)];
    acc = __builtin_amdgcn_wmma_f32_16x16x4_f32(false, a, false, b,
                                                (short)0, acc, false, false);
  }
  const int col = n0 + lm;
  const float bv = bias ? bias[col] : 0.0f;
#pragma unroll
  for (int r = 0; r < 8; ++r) {
    const int row = m0 + r + 8 * hi;
    C[(size_t)row * ldc + col] = alpha * acc[r] + bv;
  }
}

// ---------------- blocked WMMA GEMM: 256 threads = 8 waves ----------------
// Block tile: 128 rows x 64 cols. Wave w owns rows m0 = blk*128 + w*16 and the whole
// 64-col strip (4 accumulators, A-frag reused 4x). B strip [KC x 64] is staged in LDS
// (double buffered); staged with async-load-to-LDS when the toolchain exposes it.
// Requires M%128==0, N%64==0, K%4==0 (K remainder chunks handled, must be %4).
template <bool TRANS_A>
__global__ void k_wmma_gemm_blk(const float* __restrict__ A, int lda,
                                const float* __restrict__ B, int ldb,
                                float* __restrict__ C, int ldc,
                                int K, float alpha, const float* __restrict__ bias) {
  __shared__ float sB[2][KC * 64];
  const int tid = threadIdx.x;
  const int wave = tid >> 5;
  const int lane = tid & 31;
  const int lm = lane & 15;
  const int hi = lane >> 4;
  const int m0 = blockIdx.x * 128 + wave * 16;
  const int n0 = blockIdx.y * 64;
  v8f acc0 = {}, acc1 = {}, acc2 = {}, acc3 = {};

  auto stage = [&](int buf, int kc, int rows) {
    for (int e = tid; e < rows * 64; e += 256) {
      const int r = e >> 6;
      const int c = e & 63;
      const float* gp = B + (size_t)(kc + r) * ldb + n0 + c;
#if HAVE_ASYNC_LDS
      __builtin_amdgcn_global_load_async_to_lds_b32((int*)gp, (int*)&sB[buf][e], 0, 0);
#else
      sB[buf][e] = *gp;
#endif
    }
  };

  // prologue
  {
    const int rows0 = (K < KC) ? K : KC;
    stage(0, 0, rows0);
#if HAVE_ASYNC_LDS
    __builtin_amdgcn_s_wait_asynccnt(0);
#endif
    __syncthreads();
  }

  for (int kc = 0; kc < K; kc += KC) {
    const int cur = (kc / KC) & 1;
    const int rows = (K - kc < KC) ? (K - kc) : KC;
    const int knext = kc + rows;
    if (knext < K) {
      const int rowsn = (K - knext < KC) ? (K - knext) : KC;
      stage(cur ^ 1, knext, rowsn);
      // prefetch the A stream for the following chunk (emits global_prefetch_b8)
      if (!TRANS_A) __builtin_prefetch(&A[(size_t)(m0 + lm) * lda + knext], 0, 1);
      else          __builtin_prefetch(&A[(size_t)knext * lda + m0 + lm], 0, 1);
    }
    for (int k2 = 0; k2 < rows; k2 += 4) {
      const int ka = k2 + 2 * hi;
      v2f a;
      if (TRANS_A) {
        a.x = A[(size_t)(kc + ka) * lda + (m0 + lm)];
        a.y = A[(size_t)(kc + ka + 1) * lda + (m0 + lm)];
      } else {
        a.x = A[(size_t)(m0 + lm) * lda + kc + ka];
        a.y = A[(size_t)(m0 + lm) * lda + kc + ka + 1];
      }
      v2f b;
      b.x = sB[cur][ka * 64 + lm];
      b.y = sB[cur][(ka + 1) * 64 + lm];
      acc0 = __builtin_amdgcn_wmma_f32_16x16x4_f32(false, a, false, b, (short)0, acc0, false, false);
      b.x = sB[cur][ka * 64 + 16 + lm];
      b.y = sB[cur][(ka + 1) * 64 + 16 + lm];
      acc1 = __builtin_amdgcn_wmma_f32_16x16x4_f32(false, a, false, b, (short)0, acc1, false, false);
      b.x = sB[cur][ka * 64 + 32 + lm];
      b.y = sB[cur][(ka + 1) * 64 + 32 + lm];
      acc2 = __builtin_amdgcn_wmma_f32_16x16x4_f32(false, a, false, b, (short)0, acc2, false, false);
      b.x = sB[cur][ka * 64 + 48 + lm];
      b.y = sB[cur][(ka + 1) * 64 + 48 + lm];
      acc3 = __builtin_amdgcn_wmma_f32_16x16x4_f32(false, a, false, b, (short)0, acc3, false, false);
    }
#if HAVE_ASYNC_LDS
    __builtin_amdgcn_s_wait_asynccnt(0);
#endif
    __syncthreads();
  }

  const v8f accs[4] = {acc0, acc1, acc2, acc3};
#pragma unroll
  for (int nt = 0; nt < 4; ++nt) {
    const int col = n0 + nt * 16 + lm;
    const float bv = bias ? bias[col] : 0.0f;
#pragma unroll
    for (int r = 0; r < 8; ++r) {
      const int row = m0 + r + 8 * hi;
      C[(size_t)row * ldc + col] = alpha * accs[nt][r] + bv;
    }
  }
}

// ---------------- column softmax (axis 0) + entropy, in place ----------------
__global__ void k_col_softmax_ent(float* __restrict__ P, int ld, int rows,
                                  float* __restrict__ entcol) {
  const int p = blockIdx.x;
  const int t = threadIdx.x;
  __shared__ float red[256];
  float mx = -1e30f;
  for (int i = t; i < rows; i += 256) mx = fmaxf(mx, P[(size_t)i * ld + p]);
  red[t] = mx; __syncthreads();
  for (int s = 128; s > 0; s >>= 1) { if (t < s) red[t] = fmaxf(red[t], red[t + s]); __syncthreads(); }
  mx = red[0]; __syncthreads();
  float sm = 0.f;
  for (int i = t; i < rows; i += 256) sm += expf(P[(size_t)i * ld + p] - mx);
  red[t] = sm; __syncthreads();
  for (int s = 128; s > 0; s >>= 1) { if (t < s) red[t] += red[t + s]; __syncthreads(); }
  sm = red[0]; __syncthreads();
  const float logZ = logf(sm);
  const float inv = 1.0f / sm;
  float ent = 0.f;
  for (int i = t; i < rows; i += 256) {
    const float l = P[(size_t)i * ld + p] - mx;
    const float sv = expf(l) * inv;
    P[(size_t)i * ld + p] = sv;
    ent += -sv * (l - logZ);          // entr(s) = -s*ln(s)
  }
  red[t] = ent; __syncthreads();
  for (int s = 128; s > 0; s >>= 1) { if (t < s) red[t] += red[t + s]; __syncthreads(); }
  if (t == 0) entcol[p] = red[0];
}

// copy S [.,256] into B_pad cols 64..319 (ldc = 320)
__global__ void k_copyS(const float* __restrict__ S, float* __restrict__ Bp) {
  int i = blockIdx.x;
  int c = threadIdx.x;
  Bp[(size_t)i * BCOLS + 64 + c] = S[(size_t)i * P1N + c];
}

// G = [ tanh(C[:,0:64]) | C[:,64:320] | y | 0 ]  (NPK x 384)
__global__ void k_buildG(const float* __restrict__ C, const float* __restrict__ y,
                         float* __restrict__ G) {
  int i = blockIdx.x;
  for (int c = threadIdx.x; c < GCOLS; c += blockDim.x) {
    float v;
    if (c < 64)        v = tanhf(C[(size_t)i * BCOLS + c]);
    else if (c < 320)  v = C[(size_t)i * BCOLS + c];
    else if (c == 320) v = (i < NN) ? y[i] : 0.0f;
    else               v = 0.0f;
    G[(size_t)i * GCOLS + c] = v;
  }
}

// split R1 [256,384] -> x1 (scaled), A1, y1 (scaled)
__global__ void k_extract1(const float* __restrict__ R1, float* __restrict__ x1,
                           float* __restrict__ A1, float* __restrict__ y1) {
  int p = blockIdx.x;
  int c = threadIdx.x;            // 384 threads
  const float sc = (float)P1N / (float)NN;
  float v = R1[(size_t)p * GCOLS + c];
  if (c < 64)        x1[p * DD + c] = v * sc;
  else if (c < 320)  A1[p * P1N + (c - 64)] = v;
  else if (c == 320) y1[p] = v * sc;
}

__global__ void k_tanh_inplace(float* __restrict__ p, int n) {
  int i = blockIdx.x * blockDim.x + threadIdx.x;
  if (i < n) p[i] = tanhf(p[i]);
}

// G2 = [ ze2 | y1 | 0 ]   (256 x 80)
__global__ void k_buildG2(const float* __restrict__ ZE2, const float* __restrict__ y1,
                          float* __restrict__ G2) {
  int p = blockIdx.x;
  int c = threadIdx.x;            // 80 threads
  float v;
  if (c < 64)       v = ZE2[p * DD + c];
  else if (c == 64) v = y1[p];
  else              v = 0.0f;
  G2[p * G2COLS + c] = v;
}

__global__ void k_extract2(const float* __restrict__ R2, float* __restrict__ x2,
                           float* __restrict__ y2) {
  int q = blockIdx.x;
  int c = threadIdx.x;            // 80 threads
  const float sc = (float)P2N / (float)P1N;
  float v = R2[q * G2COLS + c];
  if (c < 64)       x2[q * DD + c] = v * sc;
  else if (c == 64) y2[q] = v * sc;
}

// Zr rows: [0,5000)=Z, [5000,5256)=x1, [5256,5288)=x2 ; yr likewise
__global__ void k_assemble(const float* __restrict__ Z, const float* __restrict__ x1,
                           const float* __restrict__ x2, const float* __restrict__ y,
                           const float* __restrict__ y1, const float* __restrict__ y2,
                           float* __restrict__ Zr, float* __restrict__ yr) {
  int i = blockIdx.x;             // 5288
  int d = threadIdx.x;            // 64
  float v, yv;
  if (i < NN)            { v = Z[(size_t)i * DD + d];           yv = y[i]; }
  else if (i < NN + P1N) { v = x1[(i - NN) * DD + d];           yv = y1[i - NN]; }
  else                   { v = x2[(i - NN - P1N) * DD + d];     yv = y2[i - NN - P1N]; }
  Zr[(size_t)i * DD + d] = v;
  if (d == 0) yr[i] = yv;
}

// c0[k] = b1[k] + sum_{j<6} t_enc[j]*W1[j,k]   (tx dims 6..8 are zero)
// w1sq[k] = sum_{j=1..8} W1[j,k]^2
__global__ void k_prep_decoder(const float* __restrict__ t, const float* __restrict__ W1,
                               const float* __restrict__ b1, float* __restrict__ c0,
                               float* __restrict__ w1sq) {
  int k = threadIdx.x;            // 256
  const float tv = t[0];
  float te[6];
  te[0] = tv * 1e-2f; te[1] = tv * 1e-3f; te[2] = tv * 1e-4f;
  te[3] = logf(tv * 100.f + 1.f); te[4] = logf(tv * 10.f + 1.f); te[5] = logf(tv + 1.f);
  float c = b1[k];
#pragma unroll
  for (int j = 0; j < 6; ++j) c += te[j] * W1[j * HH + k];
  c0[k] = c;
  float s = 0.f;
#pragma unroll
  for (int j = 1; j < 9; ++j) { float w = W1[j * HH + k]; s += w * w; }
  w1sq[k] = s;
}

// per-row decoder: u, g0 (d/dt), lap (sum of tx-Hessian diag over dims 1..8)
__global__ void k_decoder_rows(const float* __restrict__ Hpre, const float* __restrict__ W2,
                               const float* __restrict__ b2, const float* __restrict__ w1sq,
                               const float* __restrict__ W1row0, const float* __restrict__ yr,
                               float* __restrict__ pd, float* __restrict__ pp) {
  int i = blockIdx.x;             // 5288
  int k = threadIdx.x;            // 256
  const float hp = Hpre[(size_t)i * HH + k];
  const float h = tanhf(hp);
  const float t1 = 1.0f - h * h;
  const float w2k = W2[k];
  __shared__ float r0[256], r1[256], r2[256];
  r0[k] = h * w2k;                               // u
  r1[k] = W1row0[k] * t1 * w2k;                  // du/dt (tx dim 0)
  r2[k] = w1sq[k] * (-2.0f * h * t1) * w2k;      // laplacian over tx dims 1..8
  __syncthreads();
  for (int s = 128; s > 0; s >>= 1) {
    if (k < s) { r0[k] += r0[k + s]; r1[k] += r1[k + s]; r2[k] += r2[k + s]; }
    __syncthreads();
  }
  if (k == 0) {
    const float u = r0[0] + b2[0];
    const float e = u - yr[i];
    pd[i] = e * e;
    const float res = r1[0] - r2[0];
    pp[i] = res * res;
  }
}

__global__ void k_finalize(const float* __restrict__ pd, const float* __restrict__ pp,
                           const float* __restrict__ e1, const float* __restrict__ e2,
                           float* __restrict__ out) {
  __shared__ float red[256];
  int t = threadIdx.x;
  float sums[4];
  float a = 0.f; for (int i = t; i < MTOT; i += 256) a += pd[i];
  red[t] = a; __syncthreads();
  for (int s = 128; s > 0; s >>= 1) { if (t < s) red[t] += red[t + s]; __syncthreads(); }
  sums[0] = red[0]; __syncthreads();
  a = 0.f; for (int i = t; i < MTOT; i += 256) a += pp[i];
  red[t] = a; __syncthreads();
  for (int s = 128; s > 0; s >>= 1) { if (t < s) red[t] += red[t + s]; __syncthreads(); }
  sums[1] = red[0]; __syncthreads();
  red[t] = e1[t]; __syncthreads();
  for (int s = 128; s > 0; s >>= 1) { if (t < s) red[t] += red[t + s]; __syncthreads(); }
  sums[2] = red[0]; __syncthreads();
  red[t] = (t < P2N) ? e2[t] : 0.f; __syncthreads();
  for (int s = 128; s > 0; s >>= 1) { if (t < s) red[t] += red[t + s]; __syncthreads(); }
  sums[3] = red[0];
  if (t == 0) {
    out[0] = sums[0] / (float)MTOT + sums[1] / (float)MTOT
           + sums[2] / ((float)NN * (float)P1N)
           + sums[3] / ((float)P1N * (float)P2N);
  }
}

// ---------------- launcher ----------------
extern "C" void kernel_launch(void* const* d_in, const int* in_sizes, int n_in,
                              void* d_out, int out_size, void* d_ws, size_t ws_size,
                              hipStream_t stream) {
  const float* x0   = (const float*)d_in[0];
  const int*   adj  = (const int*)  d_in[1];
  const float* t    = (const float*)d_in[2];
  const float* y    = (const float*)d_in[3];
  const float* Wenc = (const float*)d_in[4];
  const float* Wp1  = (const float*)d_in[5];
  const float* Wp2  = (const float*)d_in[6];
  const float* We1  = (const float*)d_in[7];
  const float* We2  = (const float*)d_in[8];
  const float* W1   = (const float*)d_in[9];
  const float* b1   = (const float*)d_in[10];
  const float* W2   = (const float*)d_in[11];
  const float* b2   = (const float*)d_in[12];
  float* out = (float*)d_out;

  char* ws = (char*)d_ws;
  size_t off = 0;
  auto alloc = [&](size_t elems) -> float* {
    float* p = (float*)(ws + off);
    off += ((elems * sizeof(float) + 255) & ~(size_t)255);
    return p;
  };
  float* A_pad = alloc((size_t)NPM * NPK);    // dense adjacency, M-padded rows, K-padded cols
  float* B_pad = alloc((size_t)NPM * BCOLS);  // [E1 | S]
  float* Cbuf  = alloc((size_t)NPM * BCOLS);  // A @ [E1 | S]
  float* Z     = alloc((size_t)NPM * DD);     // encoded features (scaled)
  float* S     = alloc((size_t)NPM * P1N);    // level-1 assignment (logits -> softmax)
  float* G     = alloc((size_t)NPK * GCOLS);  // [ze1 | AS | y | 0]
  float* R1    = alloc((size_t)P1N * GCOLS);  // S^T @ G
  float* x1    = alloc((size_t)P1N * DD);
  float* A1    = alloc((size_t)P1N * P1N);
  float* y1    = alloc(P1N);
  float* S2    = alloc((size_t)P1N * P2N);
  float* E2    = alloc((size_t)P1N * DD);
  float* ZE2   = alloc((size_t)P1N * DD);
  float* G2    = alloc((size_t)P1N * G2COLS);
  float* R2    = alloc((size_t)P2N * G2COLS);
  float* x2    = alloc((size_t)P2N * DD);
  float* y2    = alloc(P2N);
  float* Zr    = alloc((size_t)MPAD * DD);
  float* yr    = alloc(MPAD);
  float* c0    = alloc(HH);
  float* w1sq  = alloc(HH);
  float* Hpre  = alloc((size_t)MPAD * HH);
  float* pd    = alloc(MTOT);
  float* pp    = alloc(MTOT);
  float* ent1  = alloc(P1N);
  float* ent2  = alloc(P2N);

  // 0) zero-init buffers whose padded regions feed WMMA tiles
  k_zero<<<4096, 256, 0, stream>>>(A_pad, (size_t)NPM * NPK);
  k_zero<<<256, 256, 0, stream>>>(Z, (size_t)NPM * DD);
  k_zero<<<256, 256, 0, stream>>>(Zr, (size_t)MPAD * DD);

  // 1) dense adjacency scatter (idempotent stores => .set(1.0) semantics)
  k_scatter_edges<<<(NEDGE + 255) / 256, 256, 0, stream>>>(adj, A_pad);

  // 2) encoder: Z = tanh(x0 @ Wenc)[:, :64] * LOG_SCALE
  k_encoder<<<NN, DD, 0, stream>>>(x0, Wenc, Z);

  // 3) level-1 logits into S: LOG_SCALE * (Z @ Wp1)   [5120 x 256]
  k_wmma_gemm_blk<false><<<dim3(NPM / 128, P1N / 64), 256, 0, stream>>>(
      Z, DD, Wp1, P1N, S, P1N, DD, LOG_SCALE, nullptr);

  // 4) softmax over axis 0 (+ entropy per column); padded rows stay 0
  k_col_softmax_ent<<<P1N, 256, 0, stream>>>(S, P1N, NN, ent1);

  // 5) B_pad[:,0:64] = Z @ We1 ; B_pad[:,64:320] = S
  k_wmma_gemm_blk<false><<<dim3(NPM / 128, DD / 64), 256, 0, stream>>>(
      Z, DD, We1, DD, B_pad, BCOLS, DD, 1.0f, nullptr);
  k_copyS<<<NPK, P1N, 0, stream>>>(S, B_pad);

  // 6) big GEMM: Cbuf = A_pad @ B_pad   (5120 x 5008 x 320)
  k_wmma_gemm_blk<false><<<dim3(NPM / 128, BCOLS / 64), 256, 0, stream>>>(
      A_pad, NPK, B_pad, BCOLS, Cbuf, BCOLS, NPK, 1.0f, nullptr);

  // 7) G = [tanh(A@E1) | A@S | y | 0]
  k_buildG<<<NPK, 256, 0, stream>>>(Cbuf, y, G);

  // 8) R1 = S^T @ G   (256 x 5008 x 384)
  k_wmma_gemm_blk<true><<<dim3(P1N / 128, GCOLS / 64), 256, 0, stream>>>(
      S, P1N, G, GCOLS, R1, GCOLS, NPK, 1.0f, nullptr);
  k_extract1<<<P1N, GCOLS, 0, stream>>>(R1, x1, A1, y1);

  // 9) level-2 logits: S2 = LOG_SCALE * (x1 @ Wp2), softmax axis 0
  k_wmma_gemm<false><<<dim3(P1N / 16, P2N / 16), 32, 0, stream>>>(
      x1, DD, Wp2, P2N, S2, P2N, DD, LOG_SCALE, nullptr);
  k_col_softmax_ent<<<P2N, 256, 0, stream>>>(S2, P2N, P1N, ent2);

  // 10) E2 = x1 @ We2 ; ZE2 = tanh(A1 @ E2)
  k_wmma_gemm_blk<false><<<dim3(P1N / 128, DD / 64), 256, 0, stream>>>(
      x1, DD, We2, DD, E2, DD, DD, 1.0f, nullptr);
  k_wmma_gemm_blk<false><<<dim3(P1N / 128, DD / 64), 256, 0, stream>>>(
      A1, P1N, E2, DD, ZE2, DD, P1N, 1.0f, nullptr);
  k_tanh_inplace<<<(P1N * DD + 255) / 256, 256, 0, stream>>>(ZE2, P1N * DD);

  // 11) R2 = S2^T @ [ze2 | y1 | 0]   (32 x 256 x 80)   [A2 = S2^T A1 S2 is dead code]
  k_buildG2<<<P1N, G2COLS, 0, stream>>>(ZE2, y1, G2);
  k_wmma_gemm<true><<<dim3(P2N / 16, G2COLS / 16), 32, 0, stream>>>(
      S2, P2N, G2, G2COLS, R2, G2COLS, P1N, 1.0f, nullptr);
  k_extract2<<<P2N, G2COLS, 0, stream>>>(R2, x2, y2);

  // 12) assemble Zr = [Z; x1; x2], yr = [y; y1; y2]
  k_assemble<<<MTOT, DD, 0, stream>>>(Z, x1, x2, y, y1, y2, Zr, yr);

  // 13) decoder prep (time encoding folded into c0) + Hpre = Zr @ W1[9:] + c0
  k_prep_decoder<<<1, HH, 0, stream>>>(t, W1, b1, c0, w1sq);
  k_wmma_gemm_blk<false><<<dim3(MPAD / 128, HH / 64), 256, 0, stream>>>(
      Zr, DD, W1 + 9 * HH, HH, Hpre, HH, DD, 1.0f, c0);

  // 14) per-row u / u_t / laplacian -> squared-error partials
  k_decoder_rows<<<MTOT, HH, 0, stream>>>(Hpre, W2, b2, w1sq, W1, yr, pd, pp);

  // 15) deterministic final reduction -> scalar loss
  k_finalize<<<1, 256, 0, stream>>>(pd, pp, ent1, ent2, out);
}